// InvDiff_26697516712382
// MI455X (gfx1250) — compile-verified
//
#include <hip/hip_runtime.h>

#define AS1 __attribute__((address_space(1)))
#define AS3 __attribute__((address_space(3)))

typedef float v4f __attribute__((ext_vector_type(4)));
typedef int   vi4 __attribute__((vector_size(16)));   // matches builtin param type

#ifndef __has_builtin
#define __has_builtin(x) 0
#endif

// ---------------------------------------------------------------------------
// CDNA5 async global->LDS copy of one 16-byte chunk (per-lane addresses).
// Tracked by ASYNCcnt; consumed after s_wait_asynccnt 0 + workgroup barrier.
// ---------------------------------------------------------------------------
__device__ __forceinline__ void async_copy_b128(const float* gsrc, float* ldst) {
#if __has_builtin(__builtin_amdgcn_global_load_async_to_lds_b128)
    __builtin_amdgcn_global_load_async_to_lds_b128(
        (AS1 vi4*)gsrc, (AS3 vi4*)ldst, /*offset=*/0, /*cpol=*/0);
#else
    // VDST = LDS byte address (low 32 bits of generic shared pointer),
    // VADDR = 64-bit global address.
    unsigned lds_addr = (unsigned)(unsigned long long)ldst;
    asm volatile("global_load_async_to_lds_b128 %0, %1, off"
                 :: "v"(lds_addr), "v"(gsrc)
                 : "memory");
#endif
}

__device__ __forceinline__ void wait_async_all() {
#if __has_builtin(__builtin_amdgcn_s_wait_asynccnt)
    __builtin_amdgcn_s_wait_asynccnt(0);
#else
    asm volatile("s_wait_asynccnt 0" ::: "memory");
#endif
}

// ---------------------------------------------------------------------------
// x: (B=64, L=4096, F=256) f32
//   d[b,i,:] = x[b,i+1,:] - x[b,i,:]          i in [0, L-1)
//   y[b,i,:] = x[b,i+1,:] - x[b,0,:]          i in [0, L-2);  y[b,L-2,:] = 0
// (the reference cumsum telescopes; no scan dependency remains)
// One block: batch b x 32 output rows. LDS tile = rows i0..i0+32 plus pivot
// row x[b,0] => 34 rows * 1KB = 34KB. 256 threads (8 wave32).
// ---------------------------------------------------------------------------
namespace {
constexpr int F    = 256;
constexpr int L    = 4096;
constexpr int B    = 64;
constexpr int ROWS = 32;                 // output rows per tile
}

__global__ __launch_bounds__(256) void InvDiff_kernel(const float* __restrict__ x,
                                                      float* __restrict__ dout,
                                                      float* __restrict__ yout) {
    __shared__ float tile[(ROWS + 2) * F];   // rows 0..32 = x[i0..i0+32], row 33 = x[b,0]

    const int b   = blockIdx.y;
    const int i0  = blockIdx.x * ROWS;
    const int tid = threadIdx.x;

    const float* xb = x + (size_t)b * L * F;

    // ---- Stage: async-load 34 rows (64 float4 each) into LDS --------------
    constexpr int NV = (ROWS + 2) * (F / 4);   // 2176 16-byte chunks
    for (int idx = tid; idx < NV; idx += 256) {
        const int r = idx >> 6;                // LDS row 0..33
        const int c = idx & 63;                // float4 column
        int grow;
        if (r <= ROWS) {
            grow = i0 + r;
            if (grow > L - 1) grow = L - 1;    // clamp (duplicate load, harmless)
        } else {
            grow = 0;                          // pivot row x[b,0]
        }
        async_copy_b128(xb + (size_t)grow * F + c * 4, &tile[r * F + c * 4]);
    }
    wait_async_all();          // my async ops done (LDS writes landed)
    __syncthreads();           // everyone's LDS writes visible

    // ---- Compute + NT stores ---------------------------------------------
    const int c4   = (tid & 63) * 4;           // float column
    const int rsub = tid >> 6;                 // 0..3; rows with stride 4

    const v4f x0 = *(const v4f*)&tile[(ROWS + 1) * F + c4];

    for (int rr = rsub; rr < ROWS; rr += 4) {
        const int i = i0 + rr;
        if (i >= L - 1) break;                 // only L-1 = 4095 output rows

        const v4f xi  = *(const v4f*)&tile[rr * F + c4];
        const v4f xi1 = *(const v4f*)&tile[(rr + 1) * F + c4];

        const v4f dv = xi1 - xi;
        v4f yv;
        if (i == L - 2) {
            yv = (v4f){0.0f, 0.0f, 0.0f, 0.0f};
        } else {
            yv = xi1 - x0;
        }

        const size_t o = ((size_t)b * (L - 1) + i) * F + c4;
        __builtin_nontemporal_store(dv, (v4f*)(dout + o));
        __builtin_nontemporal_store(yv, (v4f*)(yout + o));
    }
}

extern "C" void kernel_launch(void* const* d_in, const int* in_sizes, int n_in,
                              void* d_out, int out_size, void* d_ws, size_t ws_size,
                              hipStream_t stream) {
    (void)in_sizes; (void)n_in; (void)out_size; (void)d_ws; (void)ws_size;

    const float* x = (const float*)d_in[0];
    float* out = (float*)d_out;

    const size_t D = (size_t)B * (L - 1) * F;  // elements in d; y follows

    dim3 grid((L - 1 + ROWS - 1) / ROWS, B);   // (128, 64)
    InvDiff_kernel<<<grid, 256, 0, stream>>>(x, out, out + D);
}